// VqEmaDcrBlockPrototypeMemory_15178414424129
// MI455X (gfx1250) — compile-verified
//
#include <hip/hip_runtime.h>

// Problem constants (match reference)
#define NPROTO   512
#define NBLK     8
#define DMODEL   1024
#define DBLK     128
#define BATCH    8
#define NTOK     1024
#define TOTTOK   (BATCH * NTOK)          // 8192 flattened tokens
#define NTILES   (TOTTOK / 16)           // 512 token tiles of 16
#define NWTASK   (NTILES * NBLK)         // 4096 wave tasks
#define KSTEPS   (DBLK / 4)              // 32 wmma k-steps
#define PTILES   (NPROTO / 16)           // 32 prototype tiles

// LDS staging: chunks of 64 prototypes, double buffered, padded rows (130 f32)
#define CHUNK_P    64                    // protos per chunk
#define NCHUNK     (NPROTO / CHUNK_P)    // 8 chunks
#define PT_PER_CHK (CHUNK_P / 16)        // 4 prototype tiles per chunk
#define ROWSTRIDE  130                   // 128 + 2 dword pad (bank spread)

// Output layout: emb_st | indices | vq_loss | commitment
#define OFF_EMB  0
#define OFF_IDX  (TOTTOK * DMODEL)                  // 8388608
#define OFF_VQL  (OFF_IDX + TOTTOK * NBLK)          // 8454144
#define OFF_COM  (OFF_VQL + 1)                      // 8454145

// Workspace layout (floats): cn[4096] | partial[4096]
#define WS_CN    0
#define WS_PART  (NBLK * NPROTO)

typedef float        v2f   __attribute__((ext_vector_type(2)));
typedef float        v8f   __attribute__((ext_vector_type(8)));
typedef unsigned int u32x4 __attribute__((ext_vector_type(4)));
typedef int          i32x4 __attribute__((ext_vector_type(4)));
typedef int          i32x8 __attribute__((ext_vector_type(8)));

// ---------------- Kernel A: prototype squared norms ----------------
__global__ void vq_code_norms(const float* __restrict__ mem, float* __restrict__ ws) {
    int p = blockIdx.x * blockDim.x + threadIdx.x;      // 0..4095
    if (p >= NBLK * NPROTO) return;
    const float* row = mem + (size_t)p * DBLK;
    float s = 0.f;
    #pragma unroll 8
    for (int k = 0; k < DBLK; ++k) { float v = row[k]; s += v * v; }
    ws[WS_CN + p] = s;
}

// ---------------- TDM: DMA one 64x128 f32 chunk into padded LDS ----------------
// D# per CDNA5 ISA ch.8: group0 {count=1, lds_addr, global_addr, type=2},
// group1 {data_size=4B, pad_en, pad_interval=128dw, pad_amount=2dw,
//         tensor_dim0=128, tensor_dim1=64, tile_dim0=128, tile_dim1=64,
//         tensor_dim0_stride=128}, groups 2/3 zero (2D tensor).
__device__ __forceinline__ void tdm_load_chunk(const float* gsrc, unsigned lds_off) {
    unsigned long long ga = (unsigned long long)(uintptr_t)gsrc;
    u32x4 g0 = { 1u,                                            // count=1, user mode
                 lds_off,                                       // lds_addr (bytes)
                 (unsigned)(ga & 0xFFFFFFFFull),                // global_addr[31:0]
                 (unsigned)(((ga >> 32) & 0x1FFFFFFull) | (2u << 30)) }; // addr[56:32] | type=2
    i32x8 g1 = { (int)((2u << 16) | (1u << 20) | (6u << 22) | (1u << 25)),
                 //      data_size=4B  pad_en    interval=128dw  amount=2dw
                 (int)(128u << 16),   // bits48..63 : tensor_dim0 = 128
                 (int)(64u  << 16),   // bits80..95 : tensor_dim1 = 64
                 (int)(128u << 16),   // bits112..127: tile_dim0 = 128
                 64,                  // bits128..143: tile_dim1 = 64 (tile_dim2 = 0)
                 128,                 // bits160..191: tensor_dim0_stride = 128
                 0, 0 };
    i32x4 z4 = { 0, 0, 0, 0 };
    i32x8 z8 = { 0, 0, 0, 0, 0, 0, 0, 0 };
    __builtin_amdgcn_tensor_load_to_lds(g0, g1, z4, z4, z8, 0);
}

// ---------------- Kernel B: TDM-staged WMMA distance sweep ----------------
// 8 waves per block share one codebook block m; wave w handles token tile
// tileGroup*8 + w. Codebook streamed into LDS by the Tensor Data Mover,
// double-buffered, consumed as v_wmma_f32_16x16x4_f32 B operands via ds loads.
__global__ void __launch_bounds__(256)
vq_main(const float* __restrict__ q, const float* __restrict__ mem,
        float* __restrict__ out, float* __restrict__ ws) {
    __shared__ float ldsbuf[2][CHUNK_P * ROWSTRIDE];    // 2 x 33,280 B

    const int lane = threadIdx.x & 31;
    const int half = lane >> 4;          // 0: K even pair / M 0-7 ; 1: K odd pair / M 8-15
    const int l16  = lane & 15;
    const bool isW0 = (threadIdx.x < 32);               // wave 0 drives the TDM

    const int mblk = blockIdx.x & 7;                    // codebook block for whole WG
    const int tg   = blockIdx.x >> 3;                   // 0..63
    const int tile = tg * 8 + (threadIdx.x >> 5);       // this wave's token tile
    const int t0   = tile * 16;
    const int task = tile * 8 + mblk;                   // 0..4095 (partial-sum slot)

    const float* codeBase = mem + (size_t)mblk * NPROTO * DBLK;
    const float* cnBase   = ws + WS_CN + mblk * NPROTO;

    // ---- Load A tile (16 tokens x 128 dims) in WMMA A-layout, keep in VGPRs ----
    v2f a[KSTEPS];
    {
        const float* qrow = q + (size_t)(t0 + l16) * DMODEL + mblk * DBLK + 2 * half;
        #pragma unroll
        for (int kk = 0; kk < KSTEPS; ++kk)
            a[kk] = *(const v2f*)(qrow + 4 * kk);
    }

    float bestv[8];
    int   besti[8];
    #pragma unroll
    for (int i = 0; i < 8; ++i) { bestv[i] = 3.4e38f; besti[i] = 0; }

    // ---- TDM double-buffered pipeline over 8 chunks of 64 prototypes ----
    if (isW0)
        tdm_load_chunk(codeBase, (unsigned)(uintptr_t)&ldsbuf[0][0]);

    for (int c = 0; c < NCHUNK; ++c) {
        if (isW0) __builtin_amdgcn_s_wait_tensorcnt((short)0);  // chunk c landed
        __syncthreads();                                         // publish; buf[c^1] free
        if (isW0 && c + 1 < NCHUNK)
            tdm_load_chunk(codeBase + (size_t)(c + 1) * CHUNK_P * DBLK,
                           (unsigned)(uintptr_t)&ldsbuf[(c + 1) & 1][0]);

        const float* buf = &ldsbuf[c & 1][0];
        #pragma unroll
        for (int ptc = 0; ptc < PT_PER_CHK; ++ptc) {
            const int p_lane = (c * PT_PER_CHK + ptc) * 16 + l16; // lane's proto column
            const float* brow = buf + (ptc * 16 + l16) * ROWSTRIDE + 2 * half;

            v8f acc = {};
            #pragma unroll
            for (int kk = 0; kk < KSTEPS; ++kk) {
                v2f b = *(const v2f*)(brow + 4 * kk);             // ds_load_b64
                acc = __builtin_amdgcn_wmma_f32_16x16x4_f32(
                    false, a[kk], false, b, (short)0, acc, false, false);
            }

            const float cn_p = cnBase[p_lane];
            #pragma unroll
            for (int i = 0; i < 8; ++i) {
                float cand = cn_p - 2.0f * acc[i];
                if (cand < bestv[i]) { bestv[i] = cand; besti[i] = p_lane; }
            }
        }
    }

    // ---- Cross-lane argmin within each 16-lane half (first-hit tie-break) ----
    #pragma unroll
    for (int i = 0; i < 8; ++i) {
        float v = bestv[i]; int ix = besti[i];
        #pragma unroll
        for (int m = 1; m < 16; m <<= 1) {
            float ov = __shfl_xor(v, m, 32);
            int   oi = __shfl_xor(ix, m, 32);
            if (ov < v || (ov == v && oi < ix)) { v = ov; ix = oi; }
        }
        bestv[i] = v; besti[i] = ix;
    }

    // ---- Gather winning codes, write emb_st + indices, accumulate commitment ----
    float* outEmb = out + OFF_EMB;
    float* outIdx = out + OFF_IDX;
    float partial = 0.f;

    #pragma unroll
    for (int i = 0; i < 8; ++i) {
        #pragma unroll
        for (int h = 0; h < 2; ++h) {
            const int w     = __shfl(besti[i], h * 16, 32);  // winner for token M=i+8h
            const int token = t0 + i + 8 * h;
            const float* crow = codeBase + (size_t)w * DBLK;
            const float* qrow = q + (size_t)token * DMODEL + mblk * DBLK;
            float*       erow = outEmb + (size_t)token * DMODEL + mblk * DBLK;

            float4 cc = *(const float4*)(crow + lane * 4);
            float4 qv = *(const float4*)(qrow + lane * 4);
            *(float4*)(erow + lane * 4) = cc;

            float dx = cc.x - qv.x, dy = cc.y - qv.y, dz = cc.z - qv.z, dw = cc.w - qv.w;
            partial += dx * dx + dy * dy + dz * dz + dw * dw;

            if (lane == 0)
                outIdx[(size_t)token * NBLK + mblk] = (float)(w + mblk * NPROTO);
        }
    }

    // wave-deterministic reduction of commitment partial
    #pragma unroll
    for (int m = 16; m >= 1; m >>= 1)
        partial += __shfl_xor(partial, m, 32);
    if (lane == 0) ws[WS_PART + task] = partial;
}

// ---------------- Kernel C: deterministic final reduce ----------------
__global__ void vq_finalize(const float* __restrict__ ws, float* __restrict__ out) {
    int lane = threadIdx.x;            // 32 threads, one wave
    float s = 0.f;
    for (int i = lane; i < NWTASK; i += 32) s += ws[WS_PART + i];
    #pragma unroll
    for (int m = 16; m >= 1; m >>= 1)
        s += __shfl_xor(s, m, 32);
    if (lane == 0) {
        out[OFF_VQL] = 0.0f;
        out[OFF_COM] = s / (float)((size_t)TOTTOK * DMODEL);
    }
}

extern "C" void kernel_launch(void* const* d_in, const int* in_sizes, int n_in,
                              void* d_out, int out_size, void* d_ws, size_t ws_size,
                              hipStream_t stream) {
    const float* q   = (const float*)d_in[0];   // (8,1024,1024)
    const float* mem = (const float*)d_in[1];   // (4096,128)
    float* out = (float*)d_out;
    float* ws  = (float*)d_ws;

    vq_code_norms<<<(NBLK * NPROTO + 255) / 256, 256, 0, stream>>>(mem, ws);
    vq_main<<<NTILES / 8 * NBLK, 256, 0, stream>>>(q, mem, out, ws);
    vq_finalize<<<1, 32, 0, stream>>>(ws, out);
}